// BipartiteGraphConvolution_38457137168808
// MI455X (gfx1250) — compile-verified
//
#include <hip/hip_runtime.h>

// ---------------------------------------------------------------------------
// CDNA5 (gfx1250) bipartite graph convolution.
// fp32 WMMA (V_WMMA_F32_16X16X4_F32), wave32, LDS-staged operands,
// explicit ping-pong double-buffered k-loop (no accumulator register shuffles).
// ---------------------------------------------------------------------------

typedef float v2f __attribute__((ext_vector_type(2)));
typedef float v8f __attribute__((ext_vector_type(8)));

#define PW   288   // LDS stride (floats) per interleaved weight k-pair:
                   // 256 data + 32 pad -> lane-half bank offset 32 (conflict-free b64)
#define HSTR 132   // LDS row stride for hidden/A tiles (4-bank row skew)

__device__ __forceinline__ v8f wmma4(v2f a, v2f b, v8f c) {
  // D = A(16x4,f32) * B(4x16,f32) + C(16x16,f32)
  return __builtin_amdgcn_wmma_f32_16x16x4_f32(
      /*neg_a=*/false, a, /*neg_b=*/false, b,
      /*c_mod=*/(short)0, c, /*reuse_a=*/false, /*reuse_b=*/false);
}

// Stage W[K,128] (row-major) into k-pair-interleaved LDS:
//   dst[(k>>1)*PW + n*2 + (k&1)] = W[k*128 + n]
__device__ __forceinline__ void stage_weights(float* __restrict__ dst,
                                              const float* __restrict__ W,
                                              int K, int tid) {
  for (int i = tid; i < K * 128; i += 256) {
    const int k = i >> 7, n = i & 127;
    dst[(k >> 1) * PW + n * 2 + (k & 1)] = W[i];
  }
}

// Stage a 128-row x 128-col fp32 tile (row-major, stride HSTR) with b128 loads.
__device__ __forceinline__ void stage_tile128(float* __restrict__ dst,
                                              const float* __restrict__ src,
                                              int row0, int M, int tid) {
  for (int i = tid; i < 128 * 32; i += 256) {       // 32 float4 per row
    const int r = i >> 5, c = i & 31;
    int g = row0 + r; if (g >= M) g = M - 1;
    const float4 x = *((const float4*)(src + (size_t)g * 128) + c);
    *(float4*)&dst[r * HSTR + c * 4] = x;
  }
}

// ---------------------------------------------------------------------------
// Fully-unrolled ping-pong wave-level GEMM: acc[8] += A(16 x 4*KT) @ B(4*KT x 128).
// A: sArow[kt*4 + half*2 .. +1]   (this wave's 16-row strip, row-major LDS)
// B: sW[(kt*2+half)*PW + n*2]     (k-pair interleaved LDS)
// Operand set for step kt+1 is fetched before the WMMA chain of step kt, so
// LDS latency overlaps matrix math; two static buffers -> no register rotation.
// ---------------------------------------------------------------------------
template <int KT>
__device__ __forceinline__ void gemm_acc(v8f acc[8],
                                         const float* __restrict__ sArow,
                                         const float* __restrict__ sW,
                                         int half, int n2)
{
  v2f a0, a1;
  v2f b0[8], b1[8];

  a0 = *(const v2f*)&sArow[half * 2];
#pragma unroll
  for (int nt = 0; nt < 8; ++nt)
    b0[nt] = *(const v2f*)&sW[half * PW + nt * 32 + n2];

#pragma unroll
  for (int kt = 0; kt < KT; kt += 2) {
    // prefetch kt+1 -> (a1, b1)
    a1 = *(const v2f*)&sArow[(kt + 1) * 4 + half * 2];
#pragma unroll
    for (int nt = 0; nt < 8; ++nt)
      b1[nt] = *(const v2f*)&sW[((kt + 1) * 2 + half) * PW + nt * 32 + n2];

#pragma unroll
    for (int nt = 0; nt < 8; ++nt)
      acc[nt] = wmma4(a0, b0[nt], acc[nt]);

    // prefetch kt+2 -> (a0, b0)  (dead on final step after full unroll)
    if (kt + 2 < KT) {
      a0 = *(const v2f*)&sArow[(kt + 2) * 4 + half * 2];
#pragma unroll
      for (int nt = 0; nt < 8; ++nt)
        b0[nt] = *(const v2f*)&sW[((kt + 2) * 2 + half) * PW + nt * 32 + n2];
    }

#pragma unroll
    for (int nt = 0; nt < 8; ++nt)
      acc[nt] = wmma4(a1, b1[nt], acc[nt]);
  }
}

// ---------------------------------------------------------------------------
// OUT[M,128] = relu(X[M,K1] @ W1[K1,128] (+ B1)) @ W2[128,128] + B2
// 256 threads = 8 wave32; block tile = 128 rows; each wave: 16 rows x 128 cols.
// ---------------------------------------------------------------------------
template <int K1>
__global__ __launch_bounds__(256)
void mlp2_kernel(const float* __restrict__ X,
                 const float* __restrict__ W1, const float* __restrict__ B1,
                 const float* __restrict__ W2, const float* __restrict__ B2,
                 float* __restrict__ OUT, int M)
{
  constexpr int XSTR = K1 + 4;
  extern __shared__ float smem[];
  float* sW1 = smem;                     // (K1/2) x PW
  float* sW2 = sW1 + (K1 / 2) * PW;      // 64 x PW
  float* sX  = sW2 + 64 * PW;            // 128 x XSTR
  float* sH  = sX + 128 * XSTR;          // 128 x HSTR
  float* sB1 = sH + 128 * HSTR;          // 128
  float* sB2 = sB1 + 128;                // 128

  const int tid = threadIdx.x;
  stage_weights(sW1, W1, K1, tid);
  stage_weights(sW2, W2, 128, tid);
  if (tid < 128) { sB1[tid] = B1 ? B1[tid] : 0.0f; sB2[tid] = B2[tid]; }

  {  // stage X tile (coalesced b128)
    constexpr int C4 = K1 / 4;
    for (int i = tid; i < 128 * C4; i += 256) {
      const int r = i / C4, c = i % C4;
      int g = blockIdx.x * 128 + r; if (g >= M) g = M - 1;
      const float4 x = *((const float4*)(X + (size_t)g * K1) + c);
      *(float4*)&sX[r * XSTR + c * 4] = x;
    }
  }
  __syncthreads();

  const int wave = tid >> 5;
  const int lane = tid & 31;
  const int l16  = lane & 15;
  const int half = lane >> 4;
  const int n2   = l16 * 2;

  // ---- GEMM1: H = relu(X @ W1 + B1) ----
  v8f acc[8] = {};
  gemm_acc<K1 / 4>(acc, &sX[(wave * 16 + l16) * XSTR], sW1, half, n2);

  // bias + relu, C-layout -> row-major LDS (A source for GEMM2)
#pragma unroll
  for (int nt = 0; nt < 8; ++nt) {
#pragma unroll
    for (int v = 0; v < 8; ++v) {
      const int m = wave * 16 + half * 8 + v;
      const int n = nt * 16 + l16;
      float h = acc[nt][v] + sB1[n];
      sH[m * HSTR + n] = h > 0.0f ? h : 0.0f;
    }
  }
  __syncthreads();

  // ---- GEMM2: OUT = H @ W2 + B2 ----
  v8f acc2[8] = {};
  gemm_acc<32>(acc2, &sH[(wave * 16 + l16) * HSTR], sW2, half, n2);

#pragma unroll
  for (int nt = 0; nt < 8; ++nt) {
#pragma unroll
    for (int v = 0; v < 8; ++v) {
      const int row = blockIdx.x * 128 + wave * 16 + half * 8 + v;
      if (row < M) {
        const int n = nt * 16 + l16;
        OUT[(size_t)row * 128 + n] = acc2[nt][v] + sB2[n];
      }
    }
  }
}

// ---------------------------------------------------------------------------
// OUT[M,128] = relu([CONV|PREV][M,256] @ Wo1 + bo1) @ Wo2 + bo2
// One 128x132 LDS buffer is time-multiplexed: conv tile -> prev tile -> H tile.
// ---------------------------------------------------------------------------
__global__ __launch_bounds__(256)
void outmlp_kernel(const float* __restrict__ CONV, const float* __restrict__ PREV,
                   const float* __restrict__ W1, const float* __restrict__ B1,
                   const float* __restrict__ W2, const float* __restrict__ B2,
                   float* __restrict__ OUT, int M)
{
  extern __shared__ float smem[];
  float* sW1 = smem;                     // 128 x PW   (256 k-rows -> 128 pairs)
  float* sW2 = sW1 + 128 * PW;           // 64 x PW
  float* sA  = sW2 + 64 * PW;            // 128 x HSTR (conv -> prev -> H)
  float* sB1 = sA + 128 * HSTR;          // 128
  float* sB2 = sB1 + 128;                // 128

  const int tid = threadIdx.x;
  stage_weights(sW1, W1, 256, tid);
  stage_weights(sW2, W2, 128, tid);
  if (tid < 128) { sB1[tid] = B1[tid]; sB2[tid] = B2[tid]; }
  stage_tile128(sA, CONV, blockIdx.x * 128, M, tid);
  __syncthreads();

  const int wave = tid >> 5;
  const int lane = tid & 31;
  const int l16  = lane & 15;
  const int half = lane >> 4;
  const int n2   = l16 * 2;
  const float* sArow = &sA[(wave * 16 + l16) * HSTR];

  // ---- GEMM1 phase A: conv half (k = 0..127, Wo1 pairs 0..63) ----
  v8f acc[8] = {};
  gemm_acc<32>(acc, sArow, sW1, half, n2);

  __syncthreads();
  stage_tile128(sA, PREV, blockIdx.x * 128, M, tid);
  __syncthreads();

  // ---- GEMM1 phase B: prev half (k = 128..255, Wo1 pairs 64..127) ----
  gemm_acc<32>(acc, sArow, sW1 + 64 * PW, half, n2);
  __syncthreads();   // everyone done reading prev tile before H overwrite

  // bias + relu -> H tile (reuses sA)
#pragma unroll
  for (int nt = 0; nt < 8; ++nt) {
#pragma unroll
    for (int v = 0; v < 8; ++v) {
      const int m = wave * 16 + half * 8 + v;
      const int n = nt * 16 + l16;
      float h = acc[nt][v] + sB1[n];
      sA[m * HSTR + n] = h > 0.0f ? h : 0.0f;
    }
  }
  __syncthreads();

  // ---- GEMM2: OUT = H @ W2 + B2 ----
  v8f acc2[8] = {};
  gemm_acc<32>(acc2, sArow, sW2, half, n2);

#pragma unroll
  for (int nt = 0; nt < 8; ++nt) {
#pragma unroll
    for (int v = 0; v < 8; ++v) {
      const int row = blockIdx.x * 128 + wave * 16 + half * 8 + v;
      if (row < M) {
        const int n = nt * 16 + l16;
        OUT[(size_t)row * 128 + n] = acc2[nt][v] + sB2[n];
      }
    }
  }
}

// ---------------------------------------------------------------------------
__global__ __launch_bounds__(256)
void zero_kernel(float* __restrict__ p, long long n)
{
  long long i = (long long)blockIdx.x * blockDim.x + threadIdx.x;
  const long long stride = (long long)gridDim.x * blockDim.x;
  for (; i < n; i += stride) p[i] = 0.0f;
}

// One edge per 32 lanes; float4 gather + 4 native fp32 global atomic adds.
__global__ __launch_bounds__(256)
void edge_scatter_kernel(const int* __restrict__ EI, const float* __restrict__ EF,
                         const float* __restrict__ NB, float* __restrict__ CONV, int E)
{
  const long long t = (long long)blockIdx.x * blockDim.x + threadIdx.x;
  const int e = (int)(t >> 5);
  if (e >= E) return;
  const int lane = (int)(t & 31);
  const int src = EI[e];
  const int dst = EI[E + e];
  const float w = EF[e];
  const float4 x = *((const float4*)(NB + (size_t)src * 128) + lane);
  float* o = CONV + (size_t)dst * 128 + lane * 4;
  unsafeAtomicAdd(o + 0, w * x.x);
  unsafeAtomicAdd(o + 1, w * x.y);
  unsafeAtomicAdd(o + 2, w * x.z);
  unsafeAtomicAdd(o + 3, w * x.w);
}

// ---------------------------------------------------------------------------
extern "C" void kernel_launch(void* const* d_in, const int* in_sizes, int n_in,
                              void* d_out, int out_size, void* d_ws, size_t ws_size,
                              hipStream_t stream)
{
  (void)n_in; (void)out_size; (void)ws_size;

  const float* left  = (const float*)d_in[0];
  const int*   ei    = (const int*)  d_in[1];
  const float* ef    = (const float*)d_in[2];
  const float* right = (const float*)d_in[3];
  const float* Wl1 = (const float*)d_in[5];
  const float* bl1 = (const float*)d_in[6];
  const float* Wl2 = (const float*)d_in[7];
  const float* bl2 = (const float*)d_in[8];
  const float* Wr1 = (const float*)d_in[9];
  const float* Wr2 = (const float*)d_in[10];
  const float* br2 = (const float*)d_in[11];
  const float* Wo1 = (const float*)d_in[12];
  const float* bo1 = (const float*)d_in[13];
  const float* Wo2 = (const float*)d_in[14];
  const float* bo2 = (const float*)d_in[15];
  float* out = (float*)d_out;

  const int NL = in_sizes[0] / 64;   // left rows
  const int E  = in_sizes[2];        // edges
  const int NR = in_sizes[3] / 64;   // right rows (== scatter_out_size)

  float* nb   = (float*)d_ws;                    // neighbour features [NL,128]
  float* prev = nb + (size_t)NL * 128;           // prev features      [NR,128]
  float* conv = prev + (size_t)NR * 128;         // segment-sum accum  [NR,128]

  const size_t smem_mlp = (size_t)(32 * PW + 64 * PW + 128 * (64 + 4) +
                                   128 * HSTR + 256) * sizeof(float);
  const size_t smem_out = (size_t)(128 * PW + 64 * PW + 128 * HSTR + 256) * sizeof(float);

  hipFuncSetAttribute(reinterpret_cast<const void*>(&mlp2_kernel<64>),
                      hipFuncAttributeMaxDynamicSharedMemorySize, (int)smem_mlp);
  hipFuncSetAttribute(reinterpret_cast<const void*>(&outmlp_kernel),
                      hipFuncAttributeMaxDynamicSharedMemorySize, (int)smem_out);

  const dim3 blk(256);
  const int gl = (NL + 127) / 128;
  const int gr = (NR + 127) / 128;

  // neighbour = relu(left@Wl1+bl1)@Wl2+bl2
  mlp2_kernel<64><<<gl, blk, smem_mlp, stream>>>(left, Wl1, bl1, Wl2, bl2, nb, NL);
  // prev = relu(right@Wr1)@Wr2+br2   (no bias on first linear)
  mlp2_kernel<64><<<gr, blk, smem_mlp, stream>>>(right, Wr1, nullptr, Wr2, br2, prev, NR);

  zero_kernel<<<2048, blk, 0, stream>>>(conv, (long long)NR * 128);

  const long long tE = (long long)E * 32;
  const int ge = (int)((tE + 255) / 256);
  edge_scatter_kernel<<<ge, blk, 0, stream>>>(ei, ef, nb, conv, E);

  outmlp_kernel<<<gr, blk, smem_out, stream>>>(conv, prev, Wo1, bo1, Wo2, bo2, out, NR);
}